// NeuralMixture_68066641707629
// MI455X (gfx1250) — compile-verified
//
#include <hip/hip_runtime.h>
#include <math.h>

// Problem sizes (match reference)
#define Bb 256
#define Tt 512
#define Cc 16
#define Hh 64

// LDS pitches chosen for conflict-free wave32 WMMA fragment access:
//  WP=72:  B-frag rows k+2*hi differ by 2*72=144 ≡ 16 (mod 64) -> lane halves hit disjoint banks
//  AP1=69: A-frag addr = 69*l2 + k + 2*hi; 69 ≡ 5 (mod 64), 5 invertible -> 16 distinct banks,
//          and the +2 offset of the hi half cannot alias (5*d ≡ 2 has no |d|<16 solution)
//  AP2=133: same property (133 ≡ 5 mod 64) for K<=128 activation buffers
//  DXP=20: rows r vs r+8 differ by 160 ≡ 32 (mod 64) -> disjoint halves
#define WP  72
#define AP1 69
#define AP2 133
#define DXP 20

typedef __attribute__((ext_vector_type(2))) float v2f;
typedef __attribute__((ext_vector_type(8))) float v8f;

static __device__ __forceinline__ v8f wmma4(v2f a, v2f b, v8f c) {
  // D(16x16,f32) = A(16x4,f32) x B(4x16,f32) + C ; exact fp32 matrix op on CDNA5
  return __builtin_amdgcn_wmma_f32_16x16x4_f32(false, a, false, b, (short)0, c,
                                               false, false);
}

// One 16x16 output tile of A(16xK) @ W(Kx64), A/W in LDS.
// A layout per lane: a.x = A[m=l2][k+2*hi], a.y = A[m=l2][k+2*hi+1]
// B layout per lane: b.x = W[k+2*hi][n0+l2], b.y = W[k+2*hi+1][n0+l2]
template <int K, int APITCH>
static __device__ __forceinline__ v8f gemm_tile(const float* __restrict__ A,
                                                int acol0,
                                                const float* __restrict__ W,
                                                int n0, v8f acc, int l2, int hi) {
#pragma unroll
  for (int k = 0; k < K; k += 4) {
    v2f a, b;
    const float* ap = A + l2 * APITCH + acol0 + k + 2 * hi;
    a.x = ap[0];
    a.y = ap[1];
    const float* wp = W + (k + 2 * hi) * WP + n0 + l2;
    b.x = wp[0];
    b.y = wp[WP];
    acc = wmma4(a, b, acc);
  }
  return acc;
}

// C/D tile layout: element r of the v8f acc is row (r + 8*hi), column (n0 + l2).
template <int APITCH>
static __device__ __forceinline__ void store_tile(float* __restrict__ O, int n0,
                                                  v8f acc, int l2, int hi) {
#pragma unroll
  for (int r = 0; r < 8; ++r) O[(r + 8 * hi) * APITCH + n0 + l2] = acc[r];
}

template <int ROWS>
static __device__ __forceinline__ void load_w64(float* __restrict__ dst,
                                                const float* __restrict__ src,
                                                int tid, int nthr) {
  for (int i = tid; i < ROWS * 64; i += nthr)
    dst[(i >> 6) * WP + (i & 63)] = src[i];
}

static __device__ __forceinline__ float relu(float v) { return v > 0.f ? v : 0.f; }

// ---------------------------------------------------------------------------
// Phase A: fused embedding chain for all B*T rows.
// z_flow = Lin([t,x],W_if); z_x = Lin(spline(t),W_ic);
// z = relu(Lin([z_flow,z_x],W_emb)); z = relu(Lin(z,W_fl1)); zflow = Lin(z,W_flo)
// ---------------------------------------------------------------------------
__global__ __launch_bounds__(128) void ncde_embed_kernel(
    const float* __restrict__ x, const float* __restrict__ seq_ts,
    const float* __restrict__ coeffs, const float* __restrict__ times,
    const float* __restrict__ W_if, const float* __restrict__ b_if,
    const float* __restrict__ W_ic, const float* __restrict__ b_ic,
    const float* __restrict__ W_emb, const float* __restrict__ b_emb,
    const float* __restrict__ W_fl1, const float* __restrict__ b_fl1,
    const float* __restrict__ W_flo, const float* __restrict__ b_flo,
    float* __restrict__ zflow) {
  __shared__ float sWif[16 * WP], sWic[16 * WP], sWemb[128 * WP];
  __shared__ float sWfl1[64 * WP], sWflo[64 * WP];
  __shared__ float sb[5][64];
  __shared__ float sA[16 * AP2], sB[16 * AP2];

  const int tid = threadIdx.x, nthr = 128;
  const int lane = tid & 31, wave = tid >> 5, l2 = lane & 15, hi = lane >> 4;
  const int r0 = blockIdx.x * 16;

  load_w64<16>(sWif, W_if, tid, nthr);
  load_w64<16>(sWic, W_ic, tid, nthr);
  load_w64<128>(sWemb, W_emb, tid, nthr);
  load_w64<64>(sWfl1, W_fl1, tid, nthr);
  load_w64<64>(sWflo, W_flo, tid, nthr);
  if (tid < 64) {
    sb[0][tid] = b_if[tid];
    sb[1][tid] = b_ic[tid];
    sb[2][tid] = b_emb[tid];
    sb[3][tid] = b_fl1[tid];
    sb[4][tid] = b_flo[tid];
  }
  // Inputs: sA[:,0:16) = [t, x]; sA[:,16:32) = spline(times[t])
  for (int i = tid; i < 256; i += nthr) {
    const int m = i >> 4, c = i & 15;
    const int rr = r0 + m, b = rr / Tt, t = rr % Tt;
    sA[m * AP2 + c] = (c == 0) ? seq_ts[rr] : x[(size_t)rr * (Cc - 1) + (c - 1)];
    const int idx = (t < Tt - 1) ? t : (Tt - 2);
    const float s = times[t] - times[idx];
    const float* cf = coeffs + ((size_t)b * (Tt - 1) + idx) * (4 * Cc) + c;
    const float a0 = cf[0], b0 = cf[Cc], c0 = cf[2 * Cc], d0 = cf[3 * Cc];
    sA[m * AP2 + 16 + c] = a0 + s * (b0 + s * (c0 + s * d0));
  }
  __syncthreads();

  const int n0 = wave * 16;
  // z_flow1 (cols 0:64 of sB) and z_x (cols 64:128 of sB)
  v8f aF = {}, aX = {};
  aF = gemm_tile<16, AP2>(sA, 0, sWif, n0, aF, l2, hi);
  aX = gemm_tile<16, AP2>(sA, 16, sWic, n0, aX, l2, hi);
  const float bF = sb[0][n0 + l2], bX = sb[1][n0 + l2];
#pragma unroll
  for (int r = 0; r < 8; ++r) {
    const int m = r + 8 * hi;
    sB[m * AP2 + n0 + l2] = aF[r] + bF;
    sB[m * AP2 + 64 + n0 + l2] = aX[r] + bX;
  }
  __syncthreads();

  v8f acc = {};
  acc = gemm_tile<128, AP2>(sB, 0, sWemb, n0, acc, l2, hi);
  float bv = sb[2][n0 + l2];
#pragma unroll
  for (int r = 0; r < 8; ++r)
    sA[(r + 8 * hi) * AP2 + n0 + l2] = relu(acc[r] + bv);
  __syncthreads();

  acc = (v8f){};
  acc = gemm_tile<64, AP2>(sA, 0, sWfl1, n0, acc, l2, hi);
  bv = sb[3][n0 + l2];
#pragma unroll
  for (int r = 0; r < 8; ++r)
    sB[(r + 8 * hi) * AP2 + n0 + l2] = relu(acc[r] + bv);
  __syncthreads();

  acc = (v8f){};
  acc = gemm_tile<64, AP2>(sB, 0, sWflo, n0, acc, l2, hi);
  bv = sb[4][n0 + l2];
#pragma unroll
  for (int r = 0; r < 8; ++r)
    zflow[(size_t)(r0 + r + 8 * hi) * Hh + n0 + l2] = acc[r] + bv;
}

// ---------------------------------------------------------------------------
// Phase B: sequential RK4 CDE scan. 16 workgroups, each owns 16 batch rows
// and runs all 511 steps locally (no global sync). W_f2 held in registers as
// WMMA B-fragments (4 N-tiles x 16 K-frags per wave, 16 waves = 64 tiles).
// ---------------------------------------------------------------------------
__global__ __launch_bounds__(512, 1) void ncde_scan_kernel(
    const float* __restrict__ coeffs, const float* __restrict__ times,
    const float* __restrict__ W_ic, const float* __restrict__ b_ic,
    const float* __restrict__ W_f1, const float* __restrict__ b_f1,
    const float* __restrict__ W_f2, const float* __restrict__ b_f2,
    float* __restrict__ zcde) {
  __shared__ float sWf1[64 * WP], sWic[16 * WP];
  __shared__ float sbf1[64], sbic[64];
  __shared__ float stim[Tt];
  __shared__ float sh[16 * AP1], szb[16 * AP1], sze[16 * AP1];
  __shared__ float sk[4][16 * AP1];
  __shared__ float sdx[4][16 * DXP];

  const int tid = threadIdx.x, nthr = 512;
  const int lane = tid & 31, wave = tid >> 5, l2 = lane & 15, hi = lane >> 4;
  const int b0 = blockIdx.x * 16;

  load_w64<64>(sWf1, W_f1, tid, nthr);
  load_w64<16>(sWic, W_ic, tid, nthr);
  if (tid < 64) {
    sbf1[tid] = b_f1[tid];
    sbic[tid] = b_ic[tid];
  }
  for (int i = tid; i < Tt; i += nthr) stim[i] = times[i];

  // Register-resident W_f2 fragments: wave handles hidden units j = 4*wave+jj.
  v2f wf2[4][16];
  float bf2r[4];
#pragma unroll
  for (int jj = 0; jj < 4; ++jj) {
    const int j = wave * 4 + jj;
    const int col = 16 * j + l2;
    bf2r[jj] = b_f2[col];
#pragma unroll
    for (int kk = 0; kk < 16; ++kk) {
      const int row = 4 * kk + 2 * hi;
      wf2[jj][kk].x = W_f2[(size_t)row * (Hh * Cc) + col];
      wf2[jj][kk].y = W_f2[(size_t)(row + 1) * (Hh * Cc) + col];
    }
  }

  // z0 = Lin(X(t0), W_ic):  X(t0)[b][c] = coeffs[b][0][c] (s == 0 at knot)
  for (int i = tid; i < 256; i += nthr) {
    const int m = i >> 4, c = i & 15;
    sze[m * AP1 + c] = coeffs[((size_t)(b0 + m) * (Tt - 1)) * (4 * Cc) + c];
  }
  __syncthreads();
  if (wave < 4) {
    const int n0 = wave * 16;
    v8f acc = {};
    acc = gemm_tile<16, AP1>(sze, 0, sWic, n0, acc, l2, hi);
    const float bv = sbic[n0 + l2];
#pragma unroll
    for (int r = 0; r < 8; ++r) {
      const int m = r + 8 * hi;
      const float v = acc[r] + bv;
      szb[m * AP1 + n0 + l2] = v;
      zcde[((size_t)(b0 + m) * Tt) * Hh + n0 + l2] = v;
    }
  }
  __syncthreads();
  for (int i = tid; i < 16 * Hh; i += nthr) {
    const int m = i >> 6, h = i & 63;
    sze[m * AP1 + h] = szb[m * AP1 + h];
  }
  __syncthreads();

  for (int step = 0; step < Tt - 1; ++step) {
    const float dt = stim[step + 1] - stim[step];
    // dX/dt at the 4 RK4 eval points (spline derivative, inline)
    if (tid < 256) {
      const int m = tid >> 4, c = tid & 15, b = b0 + m;
      const float* cf = coeffs + ((size_t)b * (Tt - 1) + step) * (4 * Cc) + c;
      const float bb = cf[Cc], cc = cf[2 * Cc], dd = cf[3 * Cc];
      const float s1 = dt * (1.f / 3.f), s2 = dt * (2.f / 3.f);
      sdx[0][m * DXP + c] = bb;
      sdx[1][m * DXP + c] = bb + s1 * (2.f * cc + 3.f * dd * s1);
      sdx[2][m * DXP + c] = bb + s2 * (2.f * cc + 3.f * dd * s2);
      const int idx3 = (step + 1 < Tt - 1) ? (step + 1) : (Tt - 2);
      const float* cf3 = coeffs + ((size_t)b * (Tt - 1) + idx3) * (4 * Cc) + c;
      const float b3 = cf3[Cc], c3 = cf3[2 * Cc], d3 = cf3[3 * Cc];
      const float s3 = stim[step + 1] - stim[idx3];
      sdx[3][m * DXP + c] = b3 + s3 * (2.f * c3 + 3.f * d3 * s3);
    }
    __syncthreads();

    for (int sub = 0; sub < 4; ++sub) {
      // GEMM1: h = relu(z_eval @ W_f1 + b_f1)  (waves 0..3, one N-tile each)
      if (wave < 4) {
        const int n0 = wave * 16;
        v8f acc = {};
        acc = gemm_tile<64, AP1>(sze, 0, sWf1, n0, acc, l2, hi);
        const float bv = sbf1[n0 + l2];
#pragma unroll
        for (int r = 0; r < 8; ++r)
          sh[(r + 8 * hi) * AP1 + n0 + l2] = relu(acc[r] + bv);
      }
      __syncthreads();

      // GEMM2 + einsum: each wave computes 4 hidden units j, each is one
      // 16x16 WMMA tile of tanh(h @ W_f2 + b_f2); contract with dX over c.
      v8f acc2[4];
#pragma unroll
      for (int jj = 0; jj < 4; ++jj) {
        v8f t;
#pragma unroll
        for (int r = 0; r < 8; ++r) t[r] = bf2r[jj];
        acc2[jj] = t;
      }
#pragma unroll
      for (int kk = 0; kk < 16; ++kk) {
        v2f a;
        const float* ap = sh + l2 * AP1 + 4 * kk + 2 * hi;
        a.x = ap[0];
        a.y = ap[1];
#pragma unroll
        for (int jj = 0; jj < 4; ++jj) acc2[jj] = wmma4(a, wf2[jj][kk], acc2[jj]);
      }
      float dxr[8];
#pragma unroll
      for (int r = 0; r < 8; ++r) dxr[r] = sdx[sub][(r + 8 * hi) * DXP + l2];
      float* kdst = sk[sub];
#pragma unroll
      for (int jj = 0; jj < 4; ++jj) {
#pragma unroll
        for (int r = 0; r < 8; ++r) {
          float p = tanhf(acc2[jj][r]) * dxr[r];
          p += __shfl_xor(p, 1, 32);
          p += __shfl_xor(p, 2, 32);
          p += __shfl_xor(p, 4, 32);
          p += __shfl_xor(p, 8, 32);
          if (l2 == 0) kdst[(r + 8 * hi) * AP1 + wave * 4 + jj] = p;
        }
      }
      __syncthreads();

      // RK4 (3/8 rule) state update
      for (int i = tid; i < 16 * Hh; i += nthr) {
        const int m = i >> 6, h = i & 63;
        const int ad = m * AP1 + h;
        const float zb = szb[ad];
        if (sub == 0) {
          sze[ad] = zb + (dt * (1.f / 3.f)) * sk[0][ad];
        } else if (sub == 1) {
          sze[ad] = zb + dt * (sk[1][ad] - sk[0][ad] * (1.f / 3.f));
        } else if (sub == 2) {
          sze[ad] = zb + dt * (sk[0][ad] - sk[1][ad] + sk[2][ad]);
        } else {
          const float zn =
              zb + (sk[0][ad] + 3.f * (sk[1][ad] + sk[2][ad]) + sk[3][ad]) *
                       (dt * 0.125f);
          szb[ad] = zn;
          sze[ad] = zn;
          zcde[((size_t)(b0 + m) * Tt + (step + 1)) * Hh + h] = zn;
        }
      }
      __syncthreads();
    }
  }
}

// ---------------------------------------------------------------------------
// Phase C: head. z = Lin([zflow,zcde],W_mix) (returned!); out = Lin(relu(
// Lin(tanh(z),W_l1)), W_l2)
// ---------------------------------------------------------------------------
__global__ __launch_bounds__(128) void ncde_head_kernel(
    const float* __restrict__ zflow, const float* __restrict__ zcde,
    const float* __restrict__ W_mix, const float* __restrict__ b_mix,
    const float* __restrict__ W_l1, const float* __restrict__ b_l1,
    const float* __restrict__ W_l2, const float* __restrict__ b_l2,
    float* __restrict__ out, float* __restrict__ zout) {
  __shared__ float sWmix[128 * WP], sWl1[64 * WP];
  __shared__ float sbmix[64], sbl1[64], sWl2v[64];
  __shared__ float sA[16 * AP2], sB[16 * AP1], sC[16 * AP1];

  const int tid = threadIdx.x, nthr = 128;
  const int lane = tid & 31, wave = tid >> 5, l2 = lane & 15, hi = lane >> 4;
  const int r0 = blockIdx.x * 16;

  load_w64<128>(sWmix, W_mix, tid, nthr);
  load_w64<64>(sWl1, W_l1, tid, nthr);
  if (tid < 64) {
    sbmix[tid] = b_mix[tid];
    sbl1[tid] = b_l1[tid];
    sWl2v[tid] = W_l2[tid];  // (64,1) -> flat 64
  }
  for (int i = tid; i < 16 * 128; i += nthr) {
    const int m = i >> 7, c = i & 127;
    const size_t rr = r0 + m;
    sA[m * AP2 + c] = (c < 64) ? zflow[rr * Hh + c] : zcde[rr * Hh + (c - 64)];
  }
  __syncthreads();

  const int n0 = wave * 16;
  v8f acc = {};
  acc = gemm_tile<128, AP2>(sA, 0, sWmix, n0, acc, l2, hi);
  float bv = sbmix[n0 + l2];
#pragma unroll
  for (int r = 0; r < 8; ++r) {
    const int m = r + 8 * hi;
    const float v = acc[r] + bv;                       // returned z (pre-tanh)
    zout[(size_t)(r0 + m) * Hh + n0 + l2] = v;
    sB[m * AP1 + n0 + l2] = tanhf(v);
  }
  __syncthreads();

  acc = (v8f){};
  acc = gemm_tile<64, AP1>(sB, 0, sWl1, n0, acc, l2, hi);
  bv = sbl1[n0 + l2];
#pragma unroll
  for (int r = 0; r < 8; ++r)
    sC[(r + 8 * hi) * AP1 + n0 + l2] = relu(acc[r] + bv);
  __syncthreads();

  // 64 -> 1 dot per row: 8 lanes per row, shuffle-reduce
  const int m = tid >> 3, p = tid & 7;
  float s = 0.f;
#pragma unroll
  for (int q = 0; q < 8; ++q) s += sC[m * AP1 + p * 8 + q] * sWl2v[p * 8 + q];
  s += __shfl_xor(s, 1, 32);
  s += __shfl_xor(s, 2, 32);
  s += __shfl_xor(s, 4, 32);
  if (p == 0) out[r0 + m] = s + b_l2[0];
}

// ---------------------------------------------------------------------------
extern "C" void kernel_launch(void* const* d_in, const int* in_sizes, int n_in,
                              void* d_out, int out_size, void* d_ws,
                              size_t ws_size, hipStream_t stream) {
  const float* x = (const float*)d_in[0];
  const float* seq_ts = (const float*)d_in[1];
  // d_in[2] = seq_mask (unused by the reference forward)
  const float* coeffs = (const float*)d_in[3];
  const float* times = (const float*)d_in[4];
  const float* W_if = (const float*)d_in[5];
  const float* b_if = (const float*)d_in[6];
  const float* W_ic = (const float*)d_in[7];
  const float* b_ic = (const float*)d_in[8];
  const float* W_emb = (const float*)d_in[9];
  const float* b_emb = (const float*)d_in[10];
  const float* W_fl1 = (const float*)d_in[11];
  const float* b_fl1 = (const float*)d_in[12];
  const float* W_flo = (const float*)d_in[13];
  const float* b_flo = (const float*)d_in[14];
  const float* W_f1 = (const float*)d_in[15];
  const float* b_f1 = (const float*)d_in[16];
  const float* W_f2 = (const float*)d_in[17];
  const float* b_f2 = (const float*)d_in[18];
  const float* W_mix = (const float*)d_in[19];
  const float* b_mix = (const float*)d_in[20];
  const float* W_l1 = (const float*)d_in[21];
  const float* b_l1 = (const float*)d_in[22];
  const float* W_l2 = (const float*)d_in[23];
  const float* b_l2 = (const float*)d_in[24];

  float* zflow = (float*)d_ws;                         // B*T*H
  float* zcde = zflow + (size_t)Bb * Tt * Hh;          // B*T*H
  float* out = (float*)d_out;                          // B*T*1
  float* zout = out + (size_t)Bb * Tt;                 // B*T*H

  ncde_embed_kernel<<<(Bb * Tt) / 16, 128, 0, stream>>>(
      x, seq_ts, coeffs, times, W_if, b_if, W_ic, b_ic, W_emb, b_emb, W_fl1,
      b_fl1, W_flo, b_flo, zflow);
  ncde_scan_kernel<<<Bb / 16, 512, 0, stream>>>(coeffs, times, W_ic, b_ic,
                                                W_f1, b_f1, W_f2, b_f2, zcde);
  ncde_head_kernel<<<(Bb * Tt) / 16, 128, 0, stream>>>(
      zflow, zcde, W_mix, b_mix, W_l1, b_l1, W_l2, b_l2, out, zout);
}